// GCN_63677185130713
// MI455X (gfx1250) — compile-verified
//
#include <hip/hip_runtime.h>
#include <math.h>

typedef __attribute__((ext_vector_type(2))) float v2f;
typedef __attribute__((ext_vector_type(8))) float v8f;

// ---------------------------------------------------------------------------
// Degree / normalization
// ---------------------------------------------------------------------------
__global__ void k_init_deg(float* __restrict__ deg, int n) {
    int i = blockIdx.x * blockDim.x + threadIdx.x;
    if (i < n) deg[i] = 1.0f;  // self-loop contribution
}

__global__ void k_edge_deg(const long long* __restrict__ dst, float* __restrict__ deg,
                           long long e_cnt) {
    long long e = (long long)blockIdx.x * blockDim.x + threadIdx.x;
    if (e < e_cnt) atomicAdd(&deg[dst[e]], 1.0f);
}

__global__ void k_deg2dinv(float* __restrict__ d, int n) {
    int i = blockIdx.x * blockDim.x + threadIdx.x;
    if (i < n) {
        float v = d[i];
        d[i] = (v > 0.0f) ? rsqrtf(v) : 0.0f;
    }
}

// ---------------------------------------------------------------------------
// GEMM1: H[N x 64] = X[N x 256] @ W1[256 x 64]   (fp32 WMMA 16x16x4)
// One wave per 16x16 output tile.
// ---------------------------------------------------------------------------
__global__ void k_gemm1(const float* __restrict__ X, const float* __restrict__ W,
                        float* __restrict__ H, int M) {
    const int K = 256, NCOL = 64, NT = NCOL / 16;  // 4 N-tiles
    int wave = (blockIdx.x * blockDim.x + threadIdx.x) >> 5;
    int lane = threadIdx.x & 31;
    int mtiles = (M + 15) >> 4;
    if (wave >= mtiles * NT) return;
    int mt = wave / NT, nt = wave % NT;

    int row  = mt * 16 + (lane & 15);
    int rowc = row < M ? row : M - 1;        // clamp loads for tail tile
    int col  = nt * 16 + (lane & 15);
    int kh   = (lane >> 4) << 1;             // lanes 0-15: K=0,1; lanes 16-31: K=2,3
    const float* xr = X + (size_t)rowc * K;

    v8f acc = {};
    for (int k = 0; k < K; k += 4) {
        v2f a, b;
        a.x = xr[k + kh];
        a.y = xr[k + kh + 1];
        b.x = W[(size_t)(k + kh)     * NCOL + col];
        b.y = W[(size_t)(k + kh + 1) * NCOL + col];
        acc = __builtin_amdgcn_wmma_f32_16x16x4_f32(false, a, false, b,
                                                    (short)0, acc, false, false);
    }
    // D layout: lanes 0-15 -> M = vgpr r; lanes 16-31 -> M = 8 + r; N = lane&15
    int rb = mt * 16 + ((lane >> 4) << 3);
    float* hp = H + (size_t)rb * NCOL + col;
    if (rb + 8 <= M) {                       // full tile: branchless stores
#pragma unroll
        for (int r = 0; r < 8; ++r) hp[(size_t)r * NCOL] = acc[r];
    } else {                                 // tail tile
#pragma unroll
        for (int r = 0; r < 8; ++r)
            if (rb + r < M) hp[(size_t)r * NCOL] = acc[r];
    }
}

// ---------------------------------------------------------------------------
// Layer-1 aggregation (HID = 64)
// ---------------------------------------------------------------------------
__global__ void k_selfloop64(const float* __restrict__ dinv, const float* __restrict__ H,
                             float* __restrict__ AGG, int n) {
    int idx = blockIdx.x * blockDim.x + threadIdx.x;   // one float4 per thread
    if (idx >= n * 16) return;
    int i = idx >> 4;                                  // node
    float w = dinv[i];
    float w2 = w * w;
    float4 v = ((const float4*)H)[idx];
    v.x *= w2; v.y *= w2; v.z *= w2; v.w *= w2;
    ((float4*)AGG)[idx] = v;
}

__global__ void k_edge_agg64(const long long* __restrict__ src, const long long* __restrict__ dst,
                             const float* __restrict__ dinv, const float* __restrict__ H,
                             float* __restrict__ AGG, long long e_cnt) {
    long long e = (long long)blockIdx.x * (blockDim.x >> 5) + (threadIdx.x >> 5);
    int lane = threadIdx.x & 31;
    if (e >= e_cnt) return;
    long long s = src[e], d = dst[e];
    float w = dinv[s] * dinv[d];
    const float* hs = H + (size_t)s * 64;
    float* ad = AGG + (size_t)d * 64;
    atomicAdd(&ad[lane],      w * hs[lane]);
    atomicAdd(&ad[lane + 32], w * hs[lane + 32]);
}

// ---------------------------------------------------------------------------
// GEMM2: H2[N x 40] = relu(AGG1 + b1) @ W2[64 x 40]   (fused bias+ReLU)
// N padded to 48 (3 tiles); stores guarded to 40 cols. fp32 WMMA 16x16x4.
// ---------------------------------------------------------------------------
__global__ void k_gemm2(const float* __restrict__ A, const float* __restrict__ bias,
                        const float* __restrict__ W, float* __restrict__ H, int M) {
    const int K = 64, NOUT = 40, NT = 3;   // 48 padded cols
    int wave = (blockIdx.x * blockDim.x + threadIdx.x) >> 5;
    int lane = threadIdx.x & 31;
    int mtiles = (M + 15) >> 4;
    if (wave >= mtiles * NT) return;
    int mt = wave / NT, nt = wave % NT;

    int row  = mt * 16 + (lane & 15);
    int rowc = row < M ? row : M - 1;
    int col  = nt * 16 + (lane & 15);
    bool colok = col < NOUT;
    int kh   = (lane >> 4) << 1;
    const float* ar = A + (size_t)rowc * K;

    v8f acc = {};
    for (int k = 0; k < K; k += 4) {
        int kc = k + kh;
        v2f a, b;
        a.x = fmaxf(ar[kc]     + bias[kc],     0.0f);
        a.y = fmaxf(ar[kc + 1] + bias[kc + 1], 0.0f);
        b.x = colok ? W[(size_t)kc       * NOUT + col] : 0.0f;
        b.y = colok ? W[(size_t)(kc + 1) * NOUT + col] : 0.0f;
        acc = __builtin_amdgcn_wmma_f32_16x16x4_f32(false, a, false, b,
                                                    (short)0, acc, false, false);
    }
    int rb = mt * 16 + ((lane >> 4) << 3);
    float* hp = H + (size_t)rb * NOUT + col;
    if (colok) {
        if (rb + 8 <= M) {
#pragma unroll
            for (int r = 0; r < 8; ++r) hp[(size_t)r * NOUT] = acc[r];
        } else {
#pragma unroll
            for (int r = 0; r < 8; ++r)
                if (rb + r < M) hp[(size_t)r * NOUT] = acc[r];
        }
    }
}

// ---------------------------------------------------------------------------
// Layer-2 aggregation (NCLS = 40)
// ---------------------------------------------------------------------------
__global__ void k_selfloop40(const float* __restrict__ dinv, const float* __restrict__ H,
                             float* __restrict__ AGG, int n) {
    long long idx = (long long)blockIdx.x * blockDim.x + threadIdx.x; // one float4
    if (idx >= (long long)n * 10) return;
    int i = (int)(idx / 10);
    float w = dinv[i];
    float w2 = w * w;
    float4 v = ((const float4*)H)[idx];
    v.x *= w2; v.y *= w2; v.z *= w2; v.w *= w2;
    ((float4*)AGG)[idx] = v;
}

__global__ void k_edge_agg40(const long long* __restrict__ src, const long long* __restrict__ dst,
                             const float* __restrict__ dinv, const float* __restrict__ H,
                             float* __restrict__ AGG, long long e_cnt) {
    long long e = (long long)blockIdx.x * (blockDim.x >> 5) + (threadIdx.x >> 5);
    int lane = threadIdx.x & 31;
    if (e >= e_cnt) return;
    long long s = src[e], d = dst[e];
    float w = dinv[s] * dinv[d];
    const float* hs = H + (size_t)s * 40;
    float* ad = AGG + (size_t)d * 40;
    atomicAdd(&ad[lane], w * hs[lane]);                         // cols 0..31
    if (lane < 8) atomicAdd(&ad[lane + 32], w * hs[lane + 32]); // cols 32..39
}

// ---------------------------------------------------------------------------
// Final: out = log_softmax(AGG2 + b2), one thread per node (float4 traffic)
// ---------------------------------------------------------------------------
__global__ void k_logsoftmax40(const float* __restrict__ AGG, const float* __restrict__ b2,
                               float* __restrict__ out, int n) {
    int i = blockIdx.x * blockDim.x + threadIdx.x;
    if (i >= n) return;
    const float4* a4 = (const float4*)(AGG + (size_t)i * 40);
    const float4* b4 = (const float4*)b2;
    float v[40];
    float m = -INFINITY;
#pragma unroll
    for (int q = 0; q < 10; ++q) {
        float4 av = a4[q], bv = b4[q];
        v[4 * q + 0] = av.x + bv.x;
        v[4 * q + 1] = av.y + bv.y;
        v[4 * q + 2] = av.z + bv.z;
        v[4 * q + 3] = av.w + bv.w;
    }
#pragma unroll
    for (int c = 0; c < 40; ++c) m = fmaxf(m, v[c]);
    float ssum = 0.0f;
#pragma unroll
    for (int c = 0; c < 40; ++c) ssum += expf(v[c] - m);
    float ls = m + logf(ssum);
    float4* o4 = (float4*)(out + (size_t)i * 40);
#pragma unroll
    for (int q = 0; q < 10; ++q) {
        float4 ov;
        ov.x = v[4 * q + 0] - ls;
        ov.y = v[4 * q + 1] - ls;
        ov.z = v[4 * q + 2] - ls;
        ov.w = v[4 * q + 3] - ls;
        o4[q] = ov;
    }
}

// ---------------------------------------------------------------------------
extern "C" void kernel_launch(void* const* d_in, const int* in_sizes, int n_in,
                              void* d_out, int out_size, void* d_ws, size_t ws_size,
                              hipStream_t stream) {
    const float*     x  = (const float*)d_in[0];
    const long long* ei = (const long long*)d_in[1];
    const float*     W1 = (const float*)d_in[2];
    const float*     b1 = (const float*)d_in[3];
    const float*     W2 = (const float*)d_in[4];
    const float*     b2 = (const float*)d_in[5];
    float* out = (float*)d_out;

    const int       N = in_sizes[0] / 256;
    const long long E = in_sizes[1] / 2;
    const long long* src = ei;
    const long long* dst = ei + E;

    // workspace: dinv[N] | h1[N*64] (reused as h2[N*40]) | agg1[N*64] (reused as agg2[N*40])
    float* dinv = (float*)d_ws;
    float* h1   = dinv + N;
    float* agg1 = h1 + (size_t)N * 64;

    int gN   = (N + 255) / 256;
    int gE   = (int)((E + 255) / 256);
    int gEw  = (int)((E + 7) / 8);          // 1 wave per edge, 8 waves/block
    int mt   = (N + 15) / 16;

    // normalization coefficients
    k_init_deg<<<gN, 256, 0, stream>>>(dinv, N);
    k_edge_deg<<<gE, 256, 0, stream>>>(dst, dinv, E);
    k_deg2dinv<<<gN, 256, 0, stream>>>(dinv, N);

    // layer 1
    k_gemm1<<<(mt * 4 + 7) / 8, 256, 0, stream>>>(x, W1, h1, N);
    k_selfloop64<<<(N * 16 + 255) / 256, 256, 0, stream>>>(dinv, h1, agg1, N);
    k_edge_agg64<<<gEw, 256, 0, stream>>>(src, dst, dinv, h1, agg1, E);

    // layer 2 (bias+ReLU fused into GEMM2 A-load); h2 reuses h1 buffer
    k_gemm2<<<(mt * 3 + 7) / 8, 256, 0, stream>>>(agg1, b1, W2, h1, N);
    k_selfloop40<<<(int)(((long long)N * 10 + 255) / 256), 256, 0, stream>>>(dinv, h1, agg1, N);
    k_edge_agg40<<<gEw, 256, 0, stream>>>(src, dst, dinv, h1, agg1, E);

    // log-softmax
    k_logsoftmax40<<<gN, 256, 0, stream>>>(agg1, b2, out, N);
}